// MoE_22840636080747
// MI455X (gfx1250) — compile-verified
//
#include <hip/hip_runtime.h>

typedef __attribute__((ext_vector_type(2))) float v2f;
typedef __attribute__((ext_vector_type(8))) float v8f;

#define B_    16
#define C_    128
#define HW_   4096
#define MOE_  8
#define NE_   16
#define ISZ_  65536      // NUM_EXPERTS*H*W
#define CHW_  524288     // C*H*W
#define NOUT_ (B_*32*HW_)

// ---------------------------------------------------------------------------
// Kernel 1: clean_logits partial GEMM via V_WMMA_F32_16X16X4_F32.
// grid = (chunks, MOE_), block = 256 (8 wave32 waves). Each wave accumulates a
// 16x16 (B x NE) f32 tile over its K strip; block reduces 8 waves in LDS with
// a fixed order (deterministic) and stores a per-(chunk,m) partial tile.
// ---------------------------------------------------------------------------
__global__ __launch_bounds__(256) void moe_gemm_partials(
    const float* __restrict__ x, const float* __restrict__ wg,
    float* __restrict__ partials, int kPerBlock)
{
    const int m    = blockIdx.y;
    const int wave = threadIdx.x >> 5;
    const int lane = threadIdx.x & 31;
    const int row  = lane & 15;   // A: batch row M   | B: expert column N
    const int hi   = lane >> 4;   // 0: K={0,1}, 1: K={2,3} within each step
    const int kPerWave = kPerBlock >> 3;
    const int kbase    = blockIdx.x * kPerBlock + wave * kPerWave;

    // A[row, i] = x[row*CHW + m*ISZ + i] ;  B[i, col] = wg[m*ISZ*NE + i*NE + col]
    const float* Arow = x  + (size_t)row * CHW_ + (size_t)m * ISZ_ + kbase + hi * 2;
    const float* Bp   = wg + (size_t)m * ((size_t)ISZ_ * NE_) +
                        (size_t)(kbase + hi * 2) * NE_ + row;

    v8f acc = {};
    for (int k0 = 0; k0 < kPerWave; k0 += 4) {
        v2f a = *(const v2f*)(Arow + k0);            // A[M=row][k, k+1] (k incl. hi*2)
        v2f b;
        b.x = Bp[(size_t)k0 * NE_];                  // B[k  ][N=row]
        b.y = Bp[(size_t)(k0 + 1) * NE_];            // B[k+1][N=row]
        acc = __builtin_amdgcn_wmma_f32_16x16x4_f32(
            /*neg_a=*/false, a, /*neg_b=*/false, b,
            /*c_mod=*/(short)0, acc, /*reuse_a=*/false, /*reuse_b=*/false);
    }

    // Deterministic 8-wave reduction: each wave writes its tile to its own LDS
    // slab, then every thread sums the 8 slabs in fixed order.
    __shared__ float sh[8][256];
#pragma unroll
    for (int r = 0; r < 8; ++r)
        sh[wave][(r + hi * 8) * 16 + row] = acc[r];  // C/D layout: M=r(+8), N=row
    __syncthreads();
    float s = 0.f;
#pragma unroll
    for (int w = 0; w < 8; ++w) s += sh[w][threadIdx.x];
    partials[((size_t)blockIdx.x * MOE_ + m) * 256 + threadIdx.x] = s;
}

// ---------------------------------------------------------------------------
// Kernel 2: deterministic chunk reduction -> logits[b][m][n] (2048 floats).
// ---------------------------------------------------------------------------
__global__ void moe_reduce_logits(const float* __restrict__ partials,
                                  float* __restrict__ logits, int chunks)
{
    int t = blockIdx.x * blockDim.x + threadIdx.x;
    if (t >= B_ * MOE_ * NE_) return;
    int b = t >> 7, rem = t & 127, m = rem >> 4, n = rem & 15;
    int e = b * 16 + n;
    float s = 0.f;
    for (int c = 0; c < chunks; ++c)
        s += partials[((size_t)c * MOE_ + m) * 256 + e];
    logits[t] = s;
}

// ---------------------------------------------------------------------------
// Kernel 3: softmax + top-4 + gates + cv^2 loss. One block, 128 threads
// (thread = (b,m) pair). Loss -> out[0]; tk gates/indices -> workspace.
// ---------------------------------------------------------------------------
__global__ __launch_bounds__(128) void moe_gating(
    const float* __restrict__ logits, float* __restrict__ tkg,
    int* __restrict__ tki, float* __restrict__ out)
{
    __shared__ float gsh[B_ * MOE_ * NE_];   // gates[b][m][n]
    __shared__ float red[128];
    __shared__ float bc;

    const int t = threadIdx.x;
    for (int i = t; i < B_ * MOE_ * NE_; i += 128) gsh[i] = 0.f;
    __syncthreads();

    const int b = t >> 3, m = t & 7;
    const float* lg = logits + b * 128 + m * 16;
    float p[NE_];
    float mx = lg[0];
#pragma unroll
    for (int n = 1; n < NE_; ++n) mx = fmaxf(mx, lg[n]);
    float sum = 0.f;
#pragma unroll
    for (int n = 0; n < NE_; ++n) { p[n] = expf(lg[n] - mx); sum += p[n]; }
    const float inv = 1.f / sum;
#pragma unroll
    for (int n = 0; n < NE_; ++n) p[n] *= inv;

    // stable top-4 (strict > keeps first index on ties, matching jax top_k)
    float q[NE_];
#pragma unroll
    for (int n = 0; n < NE_; ++n) q[n] = p[n];
    int   idx[4];
    float val[4];
    float tsum = 0.f;
#pragma unroll
    for (int j = 0; j < 4; ++j) {
        int bi = 0; float bv = q[0];
#pragma unroll
        for (int n = 1; n < NE_; ++n) if (q[n] > bv) { bv = q[n]; bi = n; }
        idx[j] = bi; val[j] = bv; q[bi] = -1.f; tsum += bv;
    }
    const float ginv = 1.f / (tsum + 1e-6f);
#pragma unroll
    for (int j = 0; j < 4; ++j) {
        float g = val[j] * ginv;
        gsh[(b * 8 + m) * 16 + idx[j]] = g;
        tkg[b * 32 + m * 4 + j] = g;
        tki[b * 32 + m * 4 + j] = idx[j];
    }
    __syncthreads();

    // importance / load per (m,n); thread t = m2*16 + n2
    const int m2 = t >> 4, n2 = t & 15;
    float imp = 0.f, ld = 0.f;
#pragma unroll
    for (int bb = 0; bb < B_; ++bb) {
        float g = gsh[(bb * 8 + m2) * 16 + n2];
        imp += g;
        ld  += (g > 0.f) ? 1.f : 0.f;
    }

    float cv[2];
    const float v[2] = {imp, ld};
    for (int pass = 0; pass < 2; ++pass) {
        red[t] = v[pass];
        __syncthreads();
        for (int s2 = 64; s2 > 0; s2 >>= 1) {
            if (t < s2) red[t] += red[t + s2];
            __syncthreads();
        }
        if (t == 0) bc = red[0] * (1.f / 128.f);
        __syncthreads();
        const float mean = bc;
        const float d = v[pass] - mean;
        red[t] = d * d;
        __syncthreads();
        for (int s2 = 64; s2 > 0; s2 >>= 1) {
            if (t < s2) red[t] += red[t + s2];
            __syncthreads();
        }
        if (t == 0) bc = red[0] * (1.f / 127.f);   // ddof=1
        __syncthreads();
        const float var = bc;
        cv[pass] = var / (mean * mean + 1e-10f);
        __syncthreads();
    }
    if (t == 0) out[0] = (cv[0] + cv[1]) * 0.01f;
}

// ---------------------------------------------------------------------------
// Kernel 4: gather + scale. block = one (b, slot) pair; float4 loads,
// b32 stores (output region starts at +1 float -> 16B-misaligned).
// ---------------------------------------------------------------------------
__global__ __launch_bounds__(256) void moe_weighted(
    const float* __restrict__ x, const float* __restrict__ absf,
    const float* __restrict__ tkg, const int* __restrict__ tki,
    float* __restrict__ out)
{
    const int s = blockIdx.x;          // b*32 + (m*4 + j)
    const int b = s >> 5;
    const int j = s & 31;
    const int m = j >> 2;
    const int ch = m * 16 + tki[s];
    const float g = tkg[s];

    const float4* pa = (const float4*)(absf + ((size_t)b * C_ + ch) * HW_);
    const float4* px = (const float4*)(x    + ((size_t)b * C_ + ch) * HW_);
    float* oa = out + 1 + (size_t)s * HW_;
    float* ox = out + 1 + (size_t)NOUT_ + (size_t)s * HW_;

    for (int i = threadIdx.x; i < HW_ / 4; i += 256) {
        float4 va = pa[i], vx = px[i];
        int o = i * 4;
        oa[o + 0] = va.x * g; oa[o + 1] = va.y * g;
        oa[o + 2] = va.z * g; oa[o + 3] = va.w * g;
        ox[o + 0] = vx.x * g; ox[o + 1] = vx.y * g;
        ox[o + 2] = vx.z * g; ox[o + 3] = vx.w * g;
    }
}

extern "C" void kernel_launch(void* const* d_in, const int* in_sizes, int n_in,
                              void* d_out, int out_size, void* d_ws, size_t ws_size,
                              hipStream_t stream) {
    const float* x    = (const float*)d_in[0];
    const float* absf = (const float*)d_in[1];
    const float* wg   = (const float*)d_in[2];
    float* out = (float*)d_out;
    float* ws  = (float*)d_ws;

    // Size the K-split by available workspace (deterministic given ws_size).
    int chunks = 64;
    while (chunks > 1 &&
           ((size_t)chunks * MOE_ * 256 + 2048 + 1024) * sizeof(float) > ws_size)
        chunks >>= 1;
    const int kPerBlock = ISZ_ / chunks;   // per-wave K = kPerBlock/8 >= 4

    float* partials = ws;                                  // chunks*8*256
    float* logits   = ws + (size_t)chunks * MOE_ * 256;    // 2048
    float* tkg      = logits + B_ * MOE_ * NE_;            // 512
    int*   tki      = (int*)(tkg + B_ * 32);               // 512

    dim3 g1(chunks, MOE_);
    moe_gemm_partials<<<g1, 256, 0, stream>>>(x, wg, partials, kPerBlock);
    moe_reduce_logits<<<(B_ * MOE_ * NE_ + 255) / 256, 256, 0, stream>>>(
        partials, logits, chunks);
    moe_gating<<<1, 128, 0, stream>>>(logits, tkg, tki, out);
    moe_weighted<<<B_ * 32, 256, 0, stream>>>(x, absf, tkg, tki, out);
}